// SpectrumLoss_64931315581119
// MI455X (gfx1250) — compile-verified
//
#include <hip/hip_runtime.h>
#include <math.h>

// ---------------------------------------------------------------------------
// SpectrumLoss for MI455X (gfx1250, wave32).
//   grid = B blocks of 32 threads (1 wave == 1 batch).
//   L (packed lower-tri, 8256 f32 = 33KB) staged HBM->LDS once via the
//   Tensor Data Mover; all 10 Lanczos iterations are LDS-resident.
//   Triangular matvecs use V_WMMA_F32_16X16X4_F32 with unconditional LDS
//   loads (all packed-triangle addresses are provably in-bounds) and
//   cndmask selects only on the 4 diagonal-block tiles.
// ---------------------------------------------------------------------------

#define NSITE 128
#define TRI   8256            // 128*129/2
#define MLAN  10
#define KAPPA 0.276f

typedef float v2f  __attribute__((ext_vector_type(2)));
typedef float v8f  __attribute__((ext_vector_type(8)));
typedef unsigned int u32x4 __attribute__((ext_vector_type(4)));
typedef int   i32x4 __attribute__((ext_vector_type(4)));
typedef int   i32x8 __attribute__((ext_vector_type(8)));

__device__ __forceinline__ float wave_sum(float x) {
#pragma unroll
  for (int m = 16; m >= 1; m >>= 1) x += __shfl_xor(x, m, 32);
  return x;
}

__device__ __forceinline__ v8f wmma_step(v8f acc, float a0, float a1,
                                         float b0, float b1) {
  v2f A, B;
  A.x = a0; A.y = a1;
  B.x = b0; B.y = b1;
  return __builtin_amdgcn_wmma_f32_16x16x4_f32(
      /*neg_a=*/false, A, /*neg_b=*/false, B,
      /*c_mod=*/(short)0, acc, /*reuse_a=*/false, /*reuse_b=*/false);
}

// y = L * x   (transpose==false)  or  y = L^T * x (transpose==true)
// L packed lower-triangular in LDS: L[i][j] = Lp[i*(i+1)/2 + j], j<=i.
// A = 16x4 operator tile; B column 0 = x chunk (lanes 0/16 per B row
// striping); D column 0 (lanes 0/16) = 16 output rows.
__device__ __forceinline__ void matvec_wmma(const float* __restrict__ Lp,
                                            const float* __restrict__ x,
                                            float* __restrict__ y,
                                            int lane, bool transpose)
{
  const int  m    = lane & 15;        // N (column) index inside the tile
  const int  kb   = (lane >> 4) << 1; // this half-wave owns K = kb, kb+1
  const bool col0 = (m == 0);
  for (int rb = 0; rb < 8; ++rb) {    // 16-row output block
    v8f acc = {0.f, 0.f, 0.f, 0.f, 0.f, 0.f, 0.f, 0.f};
    const int row     = (rb << 4) + m;
    const int tri_row = (row * (row + 1)) >> 1;
    if (!transpose) {
      // interior tiles: k < 16*rb <= row, always valid -> no select
      for (int cc = 0; cc < (rb << 2); ++cc) {
        const int k0 = (cc << 2) + kb;
        const float a0 = Lp[tri_row + k0];
        const float a1 = Lp[tri_row + k0 + 1];
        const float xa = x[k0], xb = x[k0 + 1];
        acc = wmma_step(acc, a0, a1, col0 ? xa : 0.f, col0 ? xb : 0.f);
      }
      // diagonal-block tiles: unconditional in-bounds load + select
      for (int cc = (rb << 2); cc < (rb << 2) + 4; ++cc) {
        const int k0 = (cc << 2) + kb;
        const float t0 = Lp[tri_row + k0];       // <= 8255 always
        const float t1 = Lp[tri_row + k0 + 1];
        const float a0 = (k0     <= row) ? t0 : 0.f;
        const float a1 = (k0 + 1 <= row) ? t1 : 0.f;
        const float xa = x[k0], xb = x[k0 + 1];
        acc = wmma_step(acc, a0, a1, col0 ? xa : 0.f, col0 ? xb : 0.f);
      }
    } else {
      // diagonal-block tiles: A[m][k] = L[k][row], valid iff row <= k
      for (int cc = (rb << 2); cc < (rb << 2) + 4; ++cc) {
        const int k0 = (cc << 2) + kb;
        const int k1 = k0 + 1;
        const float t0 = Lp[((k0 * (k0 + 1)) >> 1) + row];  // <= 8255 always
        const float t1 = Lp[((k1 * (k1 + 1)) >> 1) + row];
        const float a0 = (row <= k0) ? t0 : 0.f;
        const float a1 = (row <= k1) ? t1 : 0.f;
        const float xa = x[k0], xb = x[k1];
        acc = wmma_step(acc, a0, a1, col0 ? xa : 0.f, col0 ? xb : 0.f);
      }
      // interior tiles: k >= 16*rb+16 > row, always valid -> no select
      for (int cc = (rb << 2) + 4; cc < 32; ++cc) {
        const int k0 = (cc << 2) + kb;
        const int k1 = k0 + 1;
        const float a0 = Lp[((k0 * (k0 + 1)) >> 1) + row];
        const float a1 = Lp[((k1 * (k1 + 1)) >> 1) + row];
        const float xa = x[k0], xb = x[k1];
        acc = wmma_step(acc, a0, a1, col0 ? xa : 0.f, col0 ? xb : 0.f);
      }
    }
    // D column 0: lane 0 -> rows rb*16+0..7, lane 16 -> rows rb*16+8..15
    if (col0) {
      const int base = (rb << 4) + ((lane >> 4) << 3);
#pragma unroll
      for (int g = 0; g < 8; ++g) y[base + g] = acc[g];
    }
  }
}

__global__ __launch_bounds__(32) void SpectrumLoss_lanczos(
    const float* __restrict__ net_out, const float* __restrict__ U1g,
    const float* __restrict__ vg, float* __restrict__ outMaxSq,
    float* __restrict__ outMinSq)
{
  __shared__ float shL[TRI];    // packed lower-triangular L
  __shared__ float shU[NSITE];  // U1 links
  __shared__ float shX[NSITE];  // v_cur
  __shared__ float shP[NSITE];  // v_prev
  __shared__ float shW[NSITE];  // D*x, later L*(L^T w)
  __shared__ float shT[NSITE];  // L^T w

  const int lane  = threadIdx.x;
  const int batch = blockIdx.x;

  // ---- TDM: DMA the 33KB packed L into LDS (one issue per wave) ----
  {
    unsigned long long ga =
        (unsigned long long)(const void*)(net_out + (size_t)batch * TRI);
    unsigned lds =
        (unsigned)(unsigned long long)(__attribute__((address_space(3))) void*)shL;
    u32x4 g0;
    g0[0] = 1u;                                    // count=1 (valid), user D#
    g0[1] = lds;                                   // lds_addr (bytes)
    g0[2] = (unsigned)ga;                          // global_addr[31:0]
    g0[3] = (unsigned)((ga >> 32) & 0x1FFFFFFull)  // global_addr[56:32]
            | (2u << 30);                          // type=2 ("image")
    i32x8 g1;
    g1[0] = (int)(2u << 16);                       // data_size=2 -> 4 bytes
    g1[1] = (int)((TRI & 0xFFFFu) << 16);          // tensor_dim0[15:0]
    g1[2] = (int)((TRI >> 16) | (1u << 16));       // dim0 hi | tensor_dim1=1
    g1[3] = (int)((unsigned)TRI << 16);            // tile_dim0 = 8256
    g1[4] = 1;                                     // tile_dim1=1, tile_dim2=0
    g1[5] = TRI;                                   // tensor_dim0_stride lo32
    g1[6] = 0;                                     // stride hi / dim1_stride
    g1[7] = 0;
    i32x4 g2 = {0, 0, 0, 0};
    i32x4 g3 = {0, 0, 0, 0};
    i32x8 g4 = {0, 0, 0, 0, 0, 0, 0, 0};
    __builtin_amdgcn_tensor_load_to_lds(g0, g1, g2, g3, g4, 0);
  }

  // ---- overlap: load U1 + v, normalize v (wave32 reduction) ----
  float vv[4];
  float ss = 0.f;
#pragma unroll
  for (int q = 0; q < 4; ++q) {
    const int i = lane + 32 * q;
    shU[i] = U1g[batch * NSITE + i];
    const float t = vg[batch * NSITE + i];
    vv[q] = t;
    ss += t * t;
  }
  ss = wave_sum(ss);
  const float invn = 1.f / sqrtf(ss);
#pragma unroll
  for (int q = 0; q < 4; ++q) {
    const int i = lane + 32 * q;
    shX[i] = vv[q] * invn;
    shP[i] = 0.f;
  }
  __builtin_amdgcn_s_wait_tensorcnt(0);  // L resident in LDS
  __syncthreads();                       // single wave: ~S_NOP

  // ---- Lanczos: 10 steps, fully LDS-resident ----
  float alph[MLAN], bet[MLAN];
  float beta = 0.f;
  for (int it = 0; it < MLAN; ++it) {
    // w = D x : x on (a,b,c) 8x8x2 lattice, 2 sites per lane
#pragma unroll
    for (int sx = 0; sx < 2; ++sx) {
      const int s = lane * 2 + sx;
      const int a = s >> 3, b = s & 7;
      const int ap = (a + 1) & 7, am = (a + 7) & 7;
      const int bp = (b + 1) & 7, bm = (b + 7) & 7;
      const float u0  = shU[((a * 8 + b) << 1)];
      const float u1  = shU[((a * 8 + b) << 1) + 1];
      const float u0m = shU[((am * 8 + b) << 1)];
      const float u1m = shU[((a * 8 + bm) << 1) + 1];
#pragma unroll
      for (int c = 0; c < 2; ++c) {
        const int idx = ((a * 8 + b) << 1) + c;
        const float hop = u0  * shX[((ap * 8 + b) << 1) + c]
                        + u1  * shX[((a * 8 + bp) << 1) + c]
                        + u0m * shX[((am * 8 + b) << 1) + c]
                        + u1m * shX[((a * 8 + bm) << 1) + c];
        shW[idx] = shX[idx] - KAPPA * hop;
      }
    }
    __syncthreads();
    matvec_wmma(shL, shW, shT, lane, true);   // t = L^T w
    __syncthreads();
    matvec_wmma(shL, shT, shW, lane, false);  // z = L t  (into shW)
    __syncthreads();

    float al = 0.f;
#pragma unroll
    for (int q = 0; q < 4; ++q) {
      const int i = lane + 32 * q;
      al += shW[i] * shX[i];
    }
    al = wave_sum(al);

    float nb = 0.f;
    float tmp[4];
#pragma unroll
    for (int q = 0; q < 4; ++q) {
      const int i = lane + 32 * q;
      const float t = shW[i] - al * shX[i] - beta * shP[i];
      tmp[q] = t;
      nb += t * t;
    }
    nb = wave_sum(nb);
    const float bn = sqrtf(nb);
    const float ib = 1.f / (bn + 1e-30f);
#pragma unroll
    for (int q = 0; q < 4; ++q) {
      const int i = lane + 32 * q;
      shP[i] = shX[i];
      shX[i] = tmp[q] * ib;
    }
    alph[it] = al;
    bet[it]  = bn;
    beta     = bn;
    __syncthreads();
  }

  // ---- 10x10 symmetric tridiagonal eigenvalues: implicit QL on lane 0 ----
  if (lane == 0) {
    float d[MLAN], e[MLAN];
#pragma unroll
    for (int i = 0; i < MLAN; ++i) d[i] = alph[i];
#pragma unroll
    for (int i = 0; i < MLAN - 1; ++i) e[i] = bet[i];
    e[MLAN - 1] = 0.f;

    for (int l = 0; l < MLAN; ++l) {
      for (int iter = 0; iter < 48; ++iter) {
        int mm = l;
        while (mm < MLAN - 1) {
          const float dd = fabsf(d[mm]) + fabsf(d[mm + 1]);
          if (fabsf(e[mm]) <= 1.1920929e-07f * dd) break;
          ++mm;
        }
        if (mm == l) break;
        float g = (d[l + 1] - d[l]) / (2.f * e[l]);
        float r = sqrtf(g * g + 1.f);
        g = d[mm] - d[l] + e[l] / (g + copysignf(r, g));
        float s = 1.f, c = 1.f, p = 0.f;
        bool under = false;
        for (int i = mm - 1; i >= l; --i) {
          const float f = s * e[i];
          const float b = c * e[i];
          r = sqrtf(f * f + g * g);
          e[i + 1] = r;
          if (r == 0.f) { d[i + 1] -= p; e[mm] = 0.f; under = true; break; }
          s = f / r;
          c = g / r;
          g = d[i + 1] - p;
          r = (d[i] - g) * s + 2.f * c * b;
          p = s * r;
          d[i + 1] = g + p;
          g = c * r - b;
        }
        if (under) continue;
        d[l] -= p;
        e[l] = g;
        e[mm] = 0.f;
      }
    }
    float mx = 0.f, mn = 3.4e38f;
#pragma unroll
    for (int i = 0; i < MLAN; ++i) {
      const float a = fabsf(d[i]);
      mx = fmaxf(mx, a);
      mn = fminf(mn, a);
    }
    outMaxSq[batch] = mx * mx;
    outMinSq[batch] = mn * mn;
  }
}

// Deterministic tree reduction: out = mean(maxsq) - mean(minsq)
__global__ __launch_bounds__(256) void SpectrumLoss_reduce(
    const float* __restrict__ mxs, const float* __restrict__ mns,
    float* __restrict__ out, int B)
{
  __shared__ float sA[256], sB[256];
  const int t = threadIdx.x;
  float a = 0.f, b = 0.f;
  for (int i = t; i < B; i += 256) { a += mxs[i]; b += mns[i]; }
  sA[t] = a;
  sB[t] = b;
  __syncthreads();
  for (int s = 128; s > 0; s >>= 1) {
    if (t < s) { sA[t] += sA[t + s]; sB[t] += sB[t + s]; }
    __syncthreads();
  }
  if (t == 0) out[0] = (sA[0] - sB[0]) / (float)B;
}

extern "C" void kernel_launch(void* const* d_in, const int* in_sizes, int n_in,
                              void* d_out, int out_size, void* d_ws, size_t ws_size,
                              hipStream_t stream) {
  const float* net_out = (const float*)d_in[0];  // (B, 8256) f32
  const float* U1      = (const float*)d_in[1];  // (B, 8, 8, 2) f32
  const float* v       = (const float*)d_in[2];  // (B, 128) f32
  float* out = (float*)d_out;
  float* ws  = (float*)d_ws;                     // [B] maxsq, [B] minsq

  const int B = in_sizes[2] / NSITE;

  SpectrumLoss_lanczos<<<dim3(B), dim3(32), 0, stream>>>(net_out, U1, v,
                                                         ws, ws + B);
  SpectrumLoss_reduce<<<dim3(1), dim3(256), 0, stream>>>(ws, ws + B, out, B);
}